// BasicGGNNCell_53008486367766
// MI455X (gfx1250) — compile-verified
//
#include <hip/hip_runtime.h>
#include <hip/hip_bf16.h>
#include <stdint.h>

#define HD 256
#define MROWS 80          // node rows per block (5 wave-tiles of 16)
#define NTHREADS 160      // 5 wave32

typedef __attribute__((ext_vector_type(2))) float v2f;
typedef __attribute__((ext_vector_type(8))) float v8f;

// Hardware f32 near-atomic add (L2 atomic unit), bypassing any CAS expansion.
__device__ __forceinline__ void atomic_add_f32_hw(float* p, float v) {
    unsigned long long a = (unsigned long long)p;
    asm volatile("global_atomic_add_f32 %0, %1, off" :: "v"(a), "v"(v) : "memory");
}

// ---------------- Kernel 0: zero the mailbox accumulator (lives in d_out) ----
__global__ void ggnn_zero_kernel(float4* __restrict__ p, long long n4) {
    long long i = (long long)blockIdx.x * blockDim.x + threadIdx.x;
    long long stride = (long long)gridDim.x * blockDim.x;
    float4 z; z.x = 0.f; z.y = 0.f; z.z = 0.f; z.w = 0.f;
    for (; i < n4; i += stride) p[i] = z;
}

// ---------------- Kernel 1: edge gather + scatter-add (mailbox sum) ---------
// One wave32 per edge; each lane handles 8 floats (two float4 chunks) of the
// 256-float row: fully coalesced 512B gathers from L2-resident h.
__global__ void ggnn_edge_kernel(const float* __restrict__ h,
                                 const float* __restrict__ relv,
                                 const int* __restrict__ esrc,
                                 const int* __restrict__ edst,
                                 const int* __restrict__ erel,
                                 float* __restrict__ red, int E) {
    int lane  = threadIdx.x & 31;
    int wave  = (blockIdx.x * blockDim.x + threadIdx.x) >> 5;
    int nwave = (gridDim.x * blockDim.x) >> 5;
    for (int e = wave; e < E; e += nwave) {
        int src = esrc[e], dst = edst[e], rel = erel[e];
        const float4* hp = (const float4*)(h + (size_t)src * HD);
        const float4* rp = (const float4*)(relv + (size_t)rel * HD);
        float* dp = red + (size_t)dst * HD;
        float4 a0 = hp[lane],      r0 = rp[lane];
        float4 a1 = hp[lane + 32], r1 = rp[lane + 32];
        a0.x += r0.x; a0.y += r0.y; a0.z += r0.z; a0.w += r0.w;
        a1.x += r1.x; a1.y += r1.y; a1.z += r1.z; a1.w += r1.w;
        int o0 = lane * 4, o1 = 128 + lane * 4;
        atomic_add_f32_hw(dp + o0 + 0, a0.x);
        atomic_add_f32_hw(dp + o0 + 1, a0.y);
        atomic_add_f32_hw(dp + o0 + 2, a0.z);
        atomic_add_f32_hw(dp + o0 + 3, a0.w);
        atomic_add_f32_hw(dp + o1 + 0, a1.x);
        atomic_add_f32_hw(dp + o1 + 1, a1.y);
        atomic_add_f32_hw(dp + o1 + 2, a1.z);
        atomic_add_f32_hw(dp + o1 + 3, a1.w);
    }
}

__device__ __forceinline__ float sigmoid_f(float x) {
    return 1.0f / (1.0f + __expf(-x));
}

// ---------------- Kernel 2: fused dual-GEMM GRU via V_WMMA_F32_16X16X4_F32 --
// Block = 160 threads (5 wave32) owning 80 node rows (wave w -> rows
// [16w,16w+16)). All waves sweep the SAME col-tile sequence, so B-fragment
// (weight) fetches are shared through L0 and per-block weight traffic
// (1.5 MB) amortizes over 80 rows: ~0.94 GB total vs 4.6 GB with 16-row
// blocks. red/h rows are staged to LDS (160 KB) behind a barrier, making the
// in-place overwrite of d_out race-free.
__global__ void __launch_bounds__(NTHREADS)
ggnn_gru_wmma_kernel(const float* __restrict__ red,
                     const float* __restrict__ h,
                     const float* __restrict__ w_ih,
                     const float* __restrict__ w_hh,
                     const float* __restrict__ b_ih,
                     const float* __restrict__ b_hh,
                     float* __restrict__ out) {
    __shared__ float sred[MROWS * HD];
    __shared__ float sh[MROWS * HD];
    const int row0 = blockIdx.x * MROWS;
    const int tid  = threadIdx.x;

    // Stage this block's 80 rows of red and h into LDS (contiguous 80KB each).
    for (int i = tid * 4; i < MROWS * HD; i += NTHREADS * 4) {
        *(float4*)(sred + i) = *(const float4*)(red + (size_t)row0 * HD + i);
        *(float4*)(sh  + i)  = *(const float4*)(h  + (size_t)row0 * HD + i);
    }
    __syncthreads();

    const int wid  = tid >> 5;           // wave id = M sub-tile (0..4)
    const int lane = tid & 31;
    const int n    = lane & 15;          // A-frag row index == B/D-frag column
    const int hi   = lane >> 4;          // half-wave select (K pair / row-group)

    const float* sredw = sred + wid * 16 * HD;   // this wave's 16 rows
    const float* shw   = sh   + wid * 16 * HD;
    const int rowW = row0 + wid * 16;

    for (int ct = 0; ct < 16; ++ct) {    // all waves in lockstep -> L0 sharing
        const int c0 = ct * 16;
        v8f air = {}, aiz = {}, ain = {}, ahr = {}, ahz = {}, ahn = {};

        const float* wr_i = w_ih + (size_t)(c0 + n) * HD;
        const float* wz_i = w_ih + (size_t)(256 + c0 + n) * HD;
        const float* wn_i = w_ih + (size_t)(512 + c0 + n) * HD;
        const float* wr_h = w_hh + (size_t)(c0 + n) * HD;
        const float* wz_h = w_hh + (size_t)(256 + c0 + n) * HD;
        const float* wn_h = w_hh + (size_t)(512 + c0 + n) * HD;

        for (int k0 = 0; k0 < HD; k0 += 4) {
            const int kk = k0 + 2 * hi;
            // A fragments: A[m][kk], A[m][kk+1] with m = lane&15
            v2f aR = *(const v2f*)(sredw + n * HD + kk);
            v2f aH = *(const v2f*)(shw   + n * HD + kk);
            // B fragments: B[k][n] = W[c0+n][k] (transposed weight rows)
            v2f bir = *(const v2f*)(wr_i + kk);
            v2f biz = *(const v2f*)(wz_i + kk);
            v2f bin = *(const v2f*)(wn_i + kk);
            v2f bhr = *(const v2f*)(wr_h + kk);
            v2f bhz = *(const v2f*)(wz_h + kk);
            v2f bhn = *(const v2f*)(wn_h + kk);

            air = __builtin_amdgcn_wmma_f32_16x16x4_f32(false, aR, false, bir, (short)0, air, false, false);
            aiz = __builtin_amdgcn_wmma_f32_16x16x4_f32(false, aR, false, biz, (short)0, aiz, false, false);
            ain = __builtin_amdgcn_wmma_f32_16x16x4_f32(false, aR, false, bin, (short)0, ain, false, false);
            ahr = __builtin_amdgcn_wmma_f32_16x16x4_f32(false, aH, false, bhr, (short)0, ahr, false, false);
            ahz = __builtin_amdgcn_wmma_f32_16x16x4_f32(false, aH, false, bhz, (short)0, ahz, false, false);
            ahn = __builtin_amdgcn_wmma_f32_16x16x4_f32(false, aH, false, bhn, (short)0, ahn, false, false);
        }

        const float bir_s = b_ih[c0 + n];
        const float biz_s = b_ih[256 + c0 + n];
        const float bin_s = b_ih[512 + c0 + n];
        const float bhr_s = b_hh[c0 + n];
        const float bhz_s = b_hh[256 + c0 + n];
        const float bhn_s = b_hh[512 + c0 + n];

#pragma unroll
        for (int v = 0; v < 8; ++v) {
            const int m = v + 8 * hi;               // D row within the 16-tile
            float ir = air[v] + bir_s, hr = ahr[v] + bhr_s;
            float iz = aiz[v] + biz_s, hz = ahz[v] + bhz_s;
            float innv = ain[v] + bin_s, hn = ahn[v] + bhn_s;
            float r  = sigmoid_f(ir + hr);
            float z  = sigmoid_f(iz + hz);
            float t  = tanhf(innv + r * hn);
            float ho = shw[m * HD + c0 + n];
            out[(size_t)(rowW + m) * HD + c0 + n] = (1.0f - z) * t + z * ho;
        }
    }
}

extern "C" void kernel_launch(void* const* d_in, const int* in_sizes, int n_in,
                              void* d_out, int out_size, void* d_ws, size_t ws_size,
                              hipStream_t stream) {
    const float* h     = (const float*)d_in[0];
    const float* relv  = (const float*)d_in[1];
    const float* w_ih  = (const float*)d_in[2];
    const float* w_hh  = (const float*)d_in[3];
    const float* b_ih  = (const float*)d_in[4];
    const float* b_hh  = (const float*)d_in[5];
    const int*   esrc  = (const int*)d_in[6];
    const int*   edst  = (const int*)d_in[7];
    const int*   erel  = (const int*)d_in[8];
    float* out = (float*)d_out;

    const int E = in_sizes[6];
    const int N = in_sizes[0] / HD;          // 50000

    // 1) zero the mailbox accumulator (stored in d_out, then rewritten in place)
    long long n4 = (long long)N * (HD / 4);
    ggnn_zero_kernel<<<2048, 256, 0, stream>>>((float4*)out, n4);

    // 2) edge gather + hardware-f32-atomic scatter (one wave per edge)
    ggnn_edge_kernel<<<4096, 256, 0, stream>>>(h, relv, esrc, edst, erel, out, E);

    // 3) fused dual-GEMM GRU with f32 WMMA; reads red from d_out via LDS stage,
    //    writes final node states back into d_out (race-free per 80-row block).
    ggnn_gru_wmma_kernel<<<N / MROWS, NTHREADS, 0, stream>>>(out, h, w_ih, w_hh,
                                                            b_ih, b_hh, out);
}